// GCNConv_50732153700353
// MI455X (gfx1250) — compile-verified
//
#include <hip/hip_runtime.h>

// GCNConv on MI455X (gfx1250, wave32).
#define IN_C   128
#define OUT_C  128
#define BATCH  4
#define NNODE  10000
#define NEDGE  160000
#define BN     (BATCH * NNODE)          // 40000 rows total
#define BE     (BATCH * NEDGE)          // 640000 edges total
#define OUT_Q  (OUT_C / 4)              // 32 float4 groups per row

typedef float v2f __attribute__((ext_vector_type(2)));
typedef float v8f __attribute__((ext_vector_type(8)));

// ---------------------------------------------------------------------------
// 1) init: zero the output accumulator (b128 stores), set deg = 1.0
// ---------------------------------------------------------------------------
__global__ void k_init(float4* __restrict__ out, float* __restrict__ deg) {
    int i = blockIdx.x * blockDim.x + threadIdx.x;
    if (i < BN * OUT_Q) out[i] = make_float4(0.0f, 0.0f, 0.0f, 0.0f);
    if (i < BN)         deg[i] = 1.0f;   // self loop contributes weight 1
}

// ---------------------------------------------------------------------------
// 2) degree accumulation: deg[b,row] += w_e   (hw f32 atomics)
// ---------------------------------------------------------------------------
__global__ void k_deg(const int* __restrict__ ei, const float* __restrict__ ew,
                      float* __restrict__ deg) {
    int i = blockIdx.x * blockDim.x + threadIdx.x;
    if (i >= BE) return;
    int b = i / NEDGE;
    int e = i - b * NEDGE;
    int row = ei[(size_t)b * 2 * NEDGE + e];          // ei[b][0][e]
    unsafeAtomicAdd(&deg[b * NNODE + row], ew[i]);
}

// ---------------------------------------------------------------------------
// 3) deg -> deg^-0.5 in place (0 if deg<=0)
// ---------------------------------------------------------------------------
__global__ void k_dinv(float* __restrict__ deg) {
    int i = blockIdx.x * blockDim.x + threadIdx.x;
    if (i >= BN) return;
    float d = deg[i];
    deg[i] = (d > 0.0f) ? rsqrtf(d) : 0.0f;
}

// ---------------------------------------------------------------------------
// 4) per-edge norm = dinv[row] * w * dinv[col]
// ---------------------------------------------------------------------------
__global__ void k_norm(const int* __restrict__ ei, const float* __restrict__ ew,
                       const float* __restrict__ dinv, float* __restrict__ norm) {
    int i = blockIdx.x * blockDim.x + threadIdx.x;
    if (i >= BE) return;
    int b = i / NEDGE;
    int e = i - b * NEDGE;
    const int* eib = ei + (size_t)b * 2 * NEDGE;
    int row = eib[e];
    int col = eib[NEDGE + e];
    norm[i] = dinv[b * NNODE + row] * ew[i] * dinv[b * NNODE + col];
}

// ---------------------------------------------------------------------------
// 5) xw = x @ W via V_WMMA_F32_16X16X4_F32 (full fp32 precision).
//    One wave -> one 16x16 output tile, K=128 in 32 steps of 4.
//    A 16x4 f32 (ISA 7.12.2): lanes 0-15 hold K={0,1}, lanes 16-31 K={2,3}.
//    B 4x16 mirrors C/D row striping. C/D: reg r = M=r / M=r+8 by lane half.
// ---------------------------------------------------------------------------
__global__ void __launch_bounds__(256) k_gemm(const float* __restrict__ X,
                                              const float* __restrict__ W,
                                              float* __restrict__ XW) {
    const int lane  = threadIdx.x & 31;
    const int ntile = threadIdx.x >> 5;        // 8 waves -> 8 N tiles
    const int mbase = blockIdx.x * 16;         // 2500 blocks, BN = 2500*16 exact
    const int nbase = ntile * 16;
    const int l15   = lane & 15;
    const int h     = lane >> 4;               // lane half
    const int h2    = h << 1;                  // K sub-offset {0,2}

    const float* xrow = X + (size_t)(mbase + l15) * IN_C;

    v8f c = {};
#pragma unroll
    for (int k0 = 0; k0 < IN_C; k0 += 4) {
        v2f a, bm;
        a.x  = xrow[k0 + h2];
        a.y  = xrow[k0 + h2 + 1];
        bm.x = W[(size_t)(k0 + h2) * OUT_C + nbase + l15];
        bm.y = W[(size_t)(k0 + h2 + 1) * OUT_C + nbase + l15];
        c = __builtin_amdgcn_wmma_f32_16x16x4_f32(
                /*neg_a=*/false, a, /*neg_b=*/false, bm,
                /*c_mod=*/(short)0, c, /*reuse_a=*/false, /*reuse_b=*/false);
    }

#pragma unroll
    for (int r = 0; r < 8; ++r) {
        int m = mbase + r + 8 * h;
        XW[(size_t)m * OUT_C + nbase + l15] = c[r];
    }
}

// ---------------------------------------------------------------------------
// 6) scatter: out[b,row,ch] += norm_e * xw[b,col,ch]
//    1 thread per (edge, float4-group): b128 gather on col (L2-resident),
//    4x hw global_atomic_add_f32 on row. 32 threads per edge.
// ---------------------------------------------------------------------------
__global__ void k_scatter(const int* __restrict__ ei, const float* __restrict__ norm,
                          const float4* __restrict__ XW, float* __restrict__ out) {
    int i = blockIdx.x * blockDim.x + threadIdx.x;   // < BE*32 = 20.48M
    if (i >= BE * OUT_Q) return;
    int q  = i & (OUT_Q - 1);                        // float4 group (0..31)
    int ed = i >> 5;                                 // edge index in [0, BE)
    int b  = ed / NEDGE;
    int e  = ed - b * NEDGE;
    const int* eib = ei + (size_t)b * 2 * NEDGE;
    int row = eib[e];
    int col = eib[NEDGE + e];
    float  nrm = norm[ed];
    float4 v   = XW[(size_t)(b * NNODE + col) * OUT_Q + q];
    float* dst = out + (size_t)(b * NNODE + row) * OUT_C + q * 4;
    unsafeAtomicAdd(dst + 0, nrm * v.x);
    unsafeAtomicAdd(dst + 1, nrm * v.y);
    unsafeAtomicAdd(dst + 2, nrm * v.z);
    unsafeAtomicAdd(dst + 3, nrm * v.w);
}

// ---------------------------------------------------------------------------
// 7) finalize: add self-loop term dinv^2 * xw (no atomics), ReLU. b128 I/O.
// ---------------------------------------------------------------------------
__global__ void k_final(const float* __restrict__ dinv, const float4* __restrict__ XW,
                        float4* __restrict__ out) {
    int i = blockIdx.x * blockDim.x + threadIdx.x;
    if (i >= BN * OUT_Q) return;
    int node = i >> 5;                  // flat b*N + n
    float di = dinv[node];
    float s  = di * di;
    float4 a = XW[i];
    float4 o = out[i];
    o.x = fmaxf(o.x + s * a.x, 0.0f);
    o.y = fmaxf(o.y + s * a.y, 0.0f);
    o.z = fmaxf(o.z + s * a.z, 0.0f);
    o.w = fmaxf(o.w + s * a.w, 0.0f);
    out[i] = o;
}

// ---------------------------------------------------------------------------
// host side
// ---------------------------------------------------------------------------
extern "C" void kernel_launch(void* const* d_in, const int* in_sizes, int n_in,
                              void* d_out, int out_size, void* d_ws, size_t ws_size,
                              hipStream_t stream) {
    const float* x  = (const float*)d_in[0];   // [B,N,128]
    const int*   ei = (const int*)d_in[1];     // [B,2,E] (int32)
    const float* ew = (const float*)d_in[2];   // [B,E]
    const float* w  = (const float*)d_in[3];   // [128,128]
    float* out = (float*)d_out;                // [B,N,128]

    // workspace layout (floats): xw | deg(->dinv) | norm  (~23.2 MB total)
    float* xw   = (float*)d_ws;                  // BN*128 = 5,120,000
    float* deg  = xw + (size_t)BN * OUT_C;       // BN     =    40,000
    float* norm = deg + BN;                      // BE     =   640,000

    const int T = 256;

    k_init<<<(BN * OUT_Q + T - 1) / T, T, 0, stream>>>((float4*)out, deg);
    k_deg<<<(BE + T - 1) / T, T, 0, stream>>>(ei, ew, deg);
    k_dinv<<<(BN + T - 1) / T, T, 0, stream>>>(deg);
    k_norm<<<(BE + T - 1) / T, T, 0, stream>>>(ei, ew, deg, norm);
    k_gemm<<<BN / 16, T, 0, stream>>>(x, w, xw);
    k_scatter<<<(BE * OUT_Q + T - 1) / T, T, 0, stream>>>(ei, norm, (const float4*)xw, out);
    k_final<<<(BN * OUT_Q + T - 1) / T, T, 0, stream>>>(deg, (const float4*)xw, (float4*)out);
}